// ColumnBlockAttention_25125558682352
// MI455X (gfx1250) — compile-verified
//
#include <hip/hip_runtime.h>
#include <math.h>

// ColumnBlockAttention for MI455X (gfx1250, wave32, WMMA).
// b=16, t=8192, e=64, BLOCK=64 -> 128 key columns per batch.
// Bandwidth-bound (writes A: 67MB + out: 33MB). Full-f32 WMMA (16x16x4) keeps
// reference precision; matrix math runs on v_wmma_f32_16x16x4_f32.

typedef float v2f __attribute__((ext_vector_type(2)));
typedef float v8f __attribute__((ext_vector_type(8)));

#define BATCH    16
#define TLEN     8192
#define EDIM     64
#define NCOLS    128        // TLEN / BLOCKSZ
#define BLOCKSZ  64
#define WAVES    4
#define ROWS_PER_WG   (WAVES * 16)   // 64 query rows per workgroup
#define KS_STRIDE 68   // 4c mod 64 distinct over 16 lanes -> conflict-free b64 reads
#define VS_STRIDE 72   // halves offset by 16 banks -> conflict-free b32 reads
#define AS_STRIDE 132  // 4m mod 64 distinct -> conflict-free b64 reads

__global__ __launch_bounds__(WAVES * 32)
void cba_kernel(const float* __restrict__ Q, const float* __restrict__ K,
                const float* __restrict__ V, float* __restrict__ Out,
                float* __restrict__ Aout) {
  extern __shared__ float smem[];
  float* Ks    = smem;                          // [NCOLS][KS_STRIDE]
  float* Vs    = Ks + NCOLS * KS_STRIDE;        // [NCOLS][VS_STRIDE]
  float* AsAll = Vs + NCOLS * VS_STRIDE;        // [WAVES][16][AS_STRIDE]

  const int tid  = threadIdx.x;
  const int wave = tid >> 5;
  const int lane = tid & 31;
  const int n    = lane & 15;   // column-within-tile / N index
  const int h    = lane >> 4;   // half-wave selector (K-pair / M-half)

  const int bi     = blockIdx.x;
  const int batch  = bi >> 7;              // 128 query tiles per batch
  const int tileq  = bi & 127;
  const int q_base = tileq * ROWS_PER_WG + wave * 16;

  const float* Qb = Q + (size_t)batch * TLEN * EDIM;
  const float* Kb = K + (size_t)batch * TLEN * EDIM;
  const float* Vb = V + (size_t)batch * TLEN * EDIM;

  // ---- cooperative gather of the 128 K/V columns into LDS ----------------
  // lanes 0..15 of a wave read one full 256B key row -> coalesced b128 loads
  for (int it = 0; it < 16; ++it) {
    int f = it * (WAVES * 32) + tid;       // 0..2047 float4 chunks
    int c = f >> 4;                        // column index 0..127
    int j = (f & 15) << 2;                 // float offset 0..60
    size_t grow = (size_t)(c * BLOCKSZ + (BLOCKSZ - 1)) * EDIM + j;
    float4 kv = *(const float4*)(Kb + grow);
    float4 vv = *(const float4*)(Vb + grow);
    *(float4*)(Ks + c * KS_STRIDE + j) = kv;
    *(float4*)(Vs + c * VS_STRIDE + j) = vv;
  }
  __syncthreads();

  // ---- load Q fragments in 16x4 A-operand layout, pre-scaled by 1/sqrt(E)
  // A layout: lane = m (0..15) | +16 for K-pair 2,3 ; VGPR0=K even, VGPR1=K odd
  const int qrow = q_base + n;
  v2f qa[16];
#pragma unroll
  for (int kk = 0; kk < 16; ++kk) {
    v2f q = *(const v2f*)(Qb + (size_t)qrow * EDIM + 4 * kk + 2 * h);
    qa[kk] = q * 0.125f;                   // 1/sqrt(64)
  }

  // ---- Z = (Q/8) * Kcols^T : 8 column tiles, K-dim 64 = 16 wmma steps ----
  v8f acc[8];
#pragma unroll
  for (int ct = 0; ct < 8; ++ct)
    acc[ct] = v8f{0.f, 0.f, 0.f, 0.f, 0.f, 0.f, 0.f, 0.f};

#pragma unroll
  for (int kk = 0; kk < 16; ++kk) {
#pragma unroll
    for (int ct = 0; ct < 8; ++ct) {
      // B layout (4x16): VGPR0 = K even row, VGPR1 = K odd; halves = K-pair
      v2f b = *(const v2f*)(Ks + (ct * 16 + n) * KS_STRIDE + 4 * kk + 2 * h);
      acc[ct] = __builtin_amdgcn_wmma_f32_16x16x4_f32(
          false, qa[kk], false, b, (short)0, acc[ct], false, false);
    }
  }

  // ---- causal mask + row softmax (C/D layout: VGPR r holds row r + 8h) ---
#pragma unroll
  for (int r = 0; r < 8; ++r) {
    int i = q_base + r + 8 * h;            // global query position
    float zv[8];
    float rmax = -3.4e38f;
#pragma unroll
    for (int ct = 0; ct < 8; ++ct) {
      int p = (ct * 16 + n) * BLOCKSZ + (BLOCKSZ - 1);  // key position
      float z = (i < p) ? -3.4e38f : acc[ct][r];
      zv[ct] = z;
      rmax = fmaxf(rmax, z);
    }
#pragma unroll
    for (int off = 1; off <= 8; off <<= 1)   // stays inside 16-lane half
      rmax = fmaxf(rmax, __shfl_xor(rmax, off, 32));
    float rsum = 0.f;
#pragma unroll
    for (int ct = 0; ct < 8; ++ct) {
      float e = (zv[ct] <= -3.0e38f) ? 0.f : __expf(zv[ct] - rmax);
      zv[ct] = e;
      rsum += e;
    }
#pragma unroll
    for (int off = 1; off <= 8; off <<= 1)
      rsum += __shfl_xor(rsum, off, 32);
    // rows before first column (i < 63) are fully masked -> A row = 0
    float inv = (i >= BLOCKSZ - 1) ? (1.0f / rsum) : 0.f;
#pragma unroll
    for (int ct = 0; ct < 8; ++ct) acc[ct][r] = zv[ct] * inv;
  }

  // ---- write A to global, stage per-wave slice in LDS for layout swap ----
  float* As = AsAll + wave * 16 * AS_STRIDE;
  float* Ag = Aout + (size_t)batch * TLEN * NCOLS;
#pragma unroll
  for (int r = 0; r < 8; ++r) {
    int m = r + 8 * h;
    int i = q_base + m;
#pragma unroll
    for (int ct = 0; ct < 8; ++ct) {
      int c = ct * 16 + n;
      float v = acc[ct][r];
      As[m * AS_STRIDE + c] = v;           // lanes 0..15 hit distinct banks
      Ag[(size_t)i * NCOLS + c] = v;       // 64B contiguous per half-wave
    }
  }

  // ---- reload A in 16x128 A-operand layout (same wave: DS stays in order)
  v2f a2[32];
#pragma unroll
  for (int kk = 0; kk < 32; ++kk)
    a2[kk] = *(const v2f*)(As + n * AS_STRIDE + 4 * kk + 2 * h);

  // ---- out = A * Vcols : 4 output tiles, K-dim 128 = 32 wmma steps ------
  v8f acc2[4];
#pragma unroll
  for (int nt = 0; nt < 4; ++nt)
    acc2[nt] = v8f{0.f, 0.f, 0.f, 0.f, 0.f, 0.f, 0.f, 0.f};

#pragma unroll
  for (int kk = 0; kk < 32; ++kk) {
#pragma unroll
    for (int nt = 0; nt < 4; ++nt) {
      v2f b;
      b.x = Vs[(4 * kk + 2 * h)     * VS_STRIDE + nt * 16 + n];
      b.y = Vs[(4 * kk + 2 * h + 1) * VS_STRIDE + nt * 16 + n];
      acc2[nt] = __builtin_amdgcn_wmma_f32_16x16x4_f32(
          false, a2[kk], false, b, (short)0, acc2[nt], false, false);
    }
  }

  // ---- store out tile [16 x 64] ------------------------------------------
  float* Og = Out + (size_t)batch * TLEN * EDIM;
#pragma unroll
  for (int r = 0; r < 8; ++r) {
    int i = q_base + r + 8 * h;
#pragma unroll
    for (int nt = 0; nt < 4; ++nt)
      Og[(size_t)i * EDIM + nt * 16 + n] = acc2[nt][r];
  }
}

extern "C" void kernel_launch(void* const* d_in, const int* in_sizes, int n_in,
                              void* d_out, int out_size, void* d_ws, size_t ws_size,
                              hipStream_t stream) {
  (void)in_sizes; (void)n_in; (void)out_size; (void)d_ws; (void)ws_size;
  const float* Q = (const float*)d_in[0];
  const float* K = (const float*)d_in[1];
  const float* V = (const float*)d_in[2];
  float* Out  = (float*)d_out;                                 // [16,8192,64]
  float* Aout = Out + (size_t)BATCH * TLEN * EDIM;             // [16,8192,128]

  dim3 grid(BATCH * (TLEN / ROWS_PER_WG));   // 16 * 128 = 2048 workgroups
  dim3 block(WAVES * 32);                    // 4 waves of 32 (wave32)
  size_t smem = (size_t)(NCOLS * KS_STRIDE + NCOLS * VS_STRIDE +
                         WAVES * 16 * AS_STRIDE) * sizeof(float);  // ~105 KB
  cba_kernel<<<grid, block, smem, stream>>>(Q, K, V, Out, Aout);
}